// SSIMLoss_71365176590857
// MI455X (gfx1250) — compile-verified
//
#include <hip/hip_runtime.h>

// SSIM loss on MI455X (gfx1250):
//  - five 7x7 box filters as banded-matrix V_WMMA_F32_16X16X4_F32 (separable V+H passes)
//  - input tiles staged to LDS by the Tensor Data Mover (tensor_load_to_lds),
//    OOB zero-fill replaces address clamping; TDM pad feature gives 33-float LDS stride
//  - elementwise SSIM map, masked accumulate, deterministic two-level reduction.

typedef float v2f __attribute__((ext_vector_type(2)));
typedef float v8f __attribute__((ext_vector_type(8)));
typedef unsigned int u32x4 __attribute__((ext_vector_type(4)));
typedef int i32x4 __attribute__((ext_vector_type(4)));
typedef int i32x8 __attribute__((ext_vector_type(8)));

#define IMG 512
#define OUTHW 506
#define PLANES 96                 // 32 batch * 3 channels
#define NTILES (PLANES * 1024)    // 32x32 tiles of 16x16 outputs per plane
#define WAVES_PER_BLOCK 2
#define NBLOCKS (NTILES / WAVES_PER_BLOCK)   // 49152
#define LSTRIDE 33                // LDS row stride in floats (TDM pad: 32dw + 1dw)

// Issue one 2D TDM load: 24 rows x 32 floats from a 512-stride plane into LDS.
// tdim0/tdim1 = remaining tensor extent from the tile origin -> OOB reads return 0.
static __device__ inline void tdm_load_2d(unsigned ldsOff, unsigned long long gaddr,
                                          int tdim0, int tdim1)
{
    u32x4 g0;
    g0.x = 1u;                                   // count=1 (valid user descriptor)
    g0.y = ldsOff;                               // lds_addr (bytes)
    g0.z = (unsigned)(gaddr & 0xffffffffu);      // global_addr[31:0]
    g0.w = ((unsigned)(gaddr >> 32) & 0x01ffffffu) | (2u << 30); // addr[56:32] | type=2

    i32x8 g1;
    g1[0] = 0x01120000;                          // data_size=4B | pad_enable | pad_interval=32dw (pad=1dw)
    g1[1] = (tdim0 & 0xffff) << 16;              // tensor_dim0[15:0]
    g1[2] = ((tdim0 >> 16) & 0xffff) | ((tdim1 & 0xffff) << 16);
    g1[3] = ((tdim1 >> 16) & 0xffff) | (32 << 16);   // tile_dim0 = 32
    g1[4] = 24;                                  // tile_dim1 = 24, tile_dim2 = 0
    g1[5] = IMG;                                 // tensor_dim0_stride = 512
    g1[6] = 0;
    g1[7] = 0;

    i32x4 z4 = {0, 0, 0, 0};
#if __has_include(<hip/amd_detail/amd_gfx1250_TDM.h>)
    i32x8 z8 = {0, 0, 0, 0, 0, 0, 0, 0};
    __builtin_amdgcn_tensor_load_to_lds(g0, g1, z4, z4, z8, 0);
#else
    __builtin_amdgcn_tensor_load_to_lds(g0, g1, z4, z4, 0);
#endif
}

__global__ __launch_bounds__(64)
void ssim_tiles(const float* __restrict__ X, const float* __restrict__ Y,
                float* __restrict__ partial)
{
    const int lane  = threadIdx.x & 31;
    const int wave  = threadIdx.x >> 5;
    const int tile  = blockIdx.x * WAVES_PER_BLOCK + wave;
    const int plane = tile >> 10;
    const int rem   = tile & 1023;
    const int r0    = (rem >> 5) << 4;   // output tile origin row
    const int c0    = (rem & 31) << 4;   // output tile origin col

    const int n     = lane & 15;         // N index (B/C/D layouts); M for A builds
    const int khalf = lane >> 4;         // K-pair half select (ISA 16x4 / 4x16 layouts)

    __shared__ float ldsX[WAVES_PER_BLOCK][24 * LSTRIDE];
    __shared__ float ldsY[WAVES_PER_BLOCK][24 * LSTRIDE];
    __shared__ float ldsV[WAVES_PER_BLOCK][5 * 16 * LSTRIDE];
    __shared__ float ldsRed[WAVES_PER_BLOCK];
    const float* Xs = ldsX[wave];
    const float* Ys = ldsY[wave];
    float* Vs = ldsV[wave];

    // ---------------- TDM: stage 24x32 input tiles of X and Y into LDS (wave-uniform D#)
    {
        const unsigned long long elemOff =
            (unsigned long long)(plane * (IMG * IMG) + r0 * IMG + c0);
        unsigned long long gx = (unsigned long long)X + 4ull * elemOff;
        unsigned long long gy = (unsigned long long)Y + 4ull * elemOff;
        unsigned xo = (unsigned)(unsigned long long)(const void*)Xs;
        unsigned yo = (unsigned)(unsigned long long)(const void*)Ys;
        int td0 = IMG - c0;              // remaining cols  -> zero-fill past right edge
        int td1 = IMG - r0;              // remaining rows  -> zero-fill past bottom edge

        // force SGPR-resident descriptor inputs
        xo  = __builtin_amdgcn_readfirstlane(xo);
        yo  = __builtin_amdgcn_readfirstlane(yo);
        td0 = __builtin_amdgcn_readfirstlane(td0);
        td1 = __builtin_amdgcn_readfirstlane(td1);
        unsigned glx = __builtin_amdgcn_readfirstlane((unsigned)gx);
        unsigned ghx = __builtin_amdgcn_readfirstlane((unsigned)(gx >> 32));
        unsigned gly = __builtin_amdgcn_readfirstlane((unsigned)gy);
        unsigned ghy = __builtin_amdgcn_readfirstlane((unsigned)(gy >> 32));

        tdm_load_2d(xo, ((unsigned long long)ghx << 32) | glx, td0, td1);
        tdm_load_2d(yo, ((unsigned long long)ghy << 32) | gly, td0, td1);
    }
    __builtin_amdgcn_s_wait_tensorcnt(0);
    asm volatile("" ::: "memory");

    // ---------------- vertical pass: V = Band(16x24) * In(24x16), 2 col blocks, 5 quantities
    v8f V[5][2];
#pragma unroll
    for (int q = 0; q < 5; ++q) { V[q][0] = (v8f){}; V[q][1] = (v8f){}; }

#pragma unroll
    for (int k = 0; k < 6; ++k) {
        const int kr0 = 4 * k + 2 * khalf;   // K row in reg .x for this lane
        const int kr1 = kr0 + 1;             // K row in reg .y
        // Band A: A[M][K] = 1 iff 0 <= K - M <= 6  (M = lane&15)
        v2f a;
        a.x = (kr0 >= n && kr0 <= n + 6) ? 1.0f : 0.0f;
        a.y = (kr1 >= n && kr1 <= n + 6) ? 1.0f : 0.0f;
#pragma unroll
        for (int cb = 0; cb < 2; ++cb) {
            const int cc = cb * 16 + n;
            const float x0 = Xs[kr0 * LSTRIDE + cc];
            const float x1 = Xs[kr1 * LSTRIDE + cc];
            const float y0 = Ys[kr0 * LSTRIDE + cc];
            const float y1 = Ys[kr1 * LSTRIDE + cc];
            const v2f bx  = {x0, x1};
            const v2f by  = {y0, y1};
            const v2f bxx = {x0 * x0, x1 * x1};
            const v2f byy = {y0 * y0, y1 * y1};
            const v2f bxy = {x0 * y0, x1 * y1};
            V[0][cb] = __builtin_amdgcn_wmma_f32_16x16x4_f32(false, a, false, bx,  (short)0, V[0][cb], false, false);
            V[1][cb] = __builtin_amdgcn_wmma_f32_16x16x4_f32(false, a, false, by,  (short)0, V[1][cb], false, false);
            V[2][cb] = __builtin_amdgcn_wmma_f32_16x16x4_f32(false, a, false, bxx, (short)0, V[2][cb], false, false);
            V[3][cb] = __builtin_amdgcn_wmma_f32_16x16x4_f32(false, a, false, byy, (short)0, V[3][cb], false, false);
            V[4][cb] = __builtin_amdgcn_wmma_f32_16x16x4_f32(false, a, false, bxy, (short)0, V[4][cb], false, false);
        }
    }

    // ---------------- stage V to LDS: C-layout -> A-layout (transpose of lane striping)
#pragma unroll
    for (int q = 0; q < 5; ++q)
#pragma unroll
        for (int cb = 0; cb < 2; ++cb)
#pragma unroll
            for (int v = 0; v < 8; ++v) {
                const int M = v + 8 * khalf;   // C/D layout: row = vgpr + 8*(lane>=16)
                Vs[(q * 16 + M) * LSTRIDE + cb * 16 + n] = V[q][cb][v];
            }
    asm volatile("s_wait_dscnt 0" ::: "memory");   // wave-private region, order store->load

    // ---------------- horizontal pass: H = V(16x24) * Band(24x16), 5 quantities
    v8f H[5];
#pragma unroll
    for (int q = 0; q < 5; ++q) H[q] = (v8f){};

#pragma unroll
    for (int k = 0; k < 6; ++k) {
        const int kc0 = 4 * k + 2 * khalf;
        const int kc1 = kc0 + 1;
        // Band B: B[K][N] = 1 iff 0 <= K - N <= 6  (N = lane&15)
        v2f b;
        b.x = (kc0 >= n && kc0 <= n + 6) ? 1.0f : 0.0f;
        b.y = (kc1 >= n && kc1 <= n + 6) ? 1.0f : 0.0f;
#pragma unroll
        for (int q = 0; q < 5; ++q) {
            v2f a;   // A[M][K]: M = lane&15; K cols 22..23 carry zero band weight
            a.x = Vs[(q * 16 + n) * LSTRIDE + kc0];
            a.y = Vs[(q * 16 + n) * LSTRIDE + kc1];
            H[q] = __builtin_amdgcn_wmma_f32_16x16x4_f32(false, a, false, b, (short)0, H[q], false, false);
        }
    }

    // ---------------- elementwise SSIM on the 8 accumulator components per lane
    const float inv_np = 1.0f / 49.0f;
    const float cov    = 49.0f / 48.0f;
    const float C1     = 1.0e-4f;   // (0.01)^2
    const float C2     = 9.0e-4f;   // (0.03)^2
    float s_local = 0.0f;
#pragma unroll
    for (int v = 0; v < 8; ++v) {
        const int M    = v + 8 * khalf;
        const int orow = r0 + M;
        const int ocol = c0 + n;
        const float ux  = H[0][v] * inv_np;
        const float uy  = H[1][v] * inv_np;
        const float vx  = cov * (H[2][v] * inv_np - ux * ux);
        const float vy  = cov * (H[3][v] * inv_np - uy * uy);
        const float vxy = cov * (H[4][v] * inv_np - ux * uy);
        const float A1  = 2.0f * ux * uy + C1;
        const float A2  = 2.0f * vxy + C2;
        const float B1  = ux * ux + uy * uy + C1;
        const float B2  = vx + vy + C2;
        const float S   = (A1 * A2) / (B1 * B2);
        if (orow < OUTHW && ocol < OUTHW) s_local += S;
    }

    // wave reduce (wave32), then block reduce to one partial per block
#pragma unroll
    for (int off = 16; off > 0; off >>= 1)
        s_local += __shfl_down(s_local, off, 32);
    if (lane == 0) ldsRed[wave] = s_local;
    __syncthreads();
    if (threadIdx.x == 0)
        partial[blockIdx.x] = ldsRed[0] + ldsRed[1];
}

__global__ __launch_bounds__(256)
void ssim_reduce(const float* __restrict__ partial, float* __restrict__ out, int nparts)
{
    __shared__ float sm[256];
    float s = 0.0f;
    for (int i = threadIdx.x; i < nparts; i += 256) s += partial[i];
    sm[threadIdx.x] = s;
    __syncthreads();
    for (int stride = 128; stride > 0; stride >>= 1) {
        if ((int)threadIdx.x < stride) sm[threadIdx.x] += sm[threadIdx.x + stride];
        __syncthreads();
    }
    if (threadIdx.x == 0)
        out[0] = 1.0f - sm[0] * (1.0f / 24579456.0f);   // 506*506*96
}

extern "C" void kernel_launch(void* const* d_in, const int* in_sizes, int n_in,
                              void* d_out, int out_size, void* d_ws, size_t ws_size,
                              hipStream_t stream) {
    const float* X = (const float*)d_in[0];
    const float* Y = (const float*)d_in[1];
    float* out     = (float*)d_out;
    float* partial = (float*)d_ws;          // NBLOCKS floats (~192 KB)

    ssim_tiles<<<NBLOCKS, 64, 0, stream>>>(X, Y, partial);
    ssim_reduce<<<1, 256, 0, stream>>>(partial, out, NBLOCKS);
}